// loss_concentration_89867895702075
// MI455X (gfx1250) — compile-verified
//
#include <hip/hip_runtime.h>
#include <hip/hip_bf16.h>
#include <math.h>

// ---- problem constants (from reference: softmask [16, 64, 240, 320] fp32) ----
#define HW        76800        // 240*320 elements per slice (contiguous)
#define NSLICE    1024         // 16*64 slices
#define TPB       256          // 8 wave32s per block
#define V4_PER_SL (HW / 4)     // 19200 float4 per slice
#define ITERS     (V4_PER_SL / TPB)  // 75 float4 loads per thread

typedef __attribute__((ext_vector_type(4))) float v4f;
typedef __attribute__((ext_vector_type(2))) float v2f;
typedef __attribute__((ext_vector_type(8))) float v8f;

// Kernel 1: one block per (b,k) slice. Streams the slice once with NT b128
// loads, accumulates sum and sum-of-squares in 4 independent fp32 chains per
// thread, reduces wave32 -> block, writes per-slice unbiased variance.
__global__ __launch_bounds__(TPB)
void slice_var_kernel(const float* __restrict__ x, float* __restrict__ vars)
{
    const int slice = blockIdx.x;
    const v4f* __restrict__ p = (const v4f*)(x + (size_t)slice * HW);
    const int t = threadIdx.x;

    float s0 = 0.f, s1 = 0.f, s2 = 0.f, s3 = 0.f;   // sums
    float q0 = 0.f, q1 = 0.f, q2 = 0.f, q3 = 0.f;   // sums of squares

#pragma unroll 5
    for (int i = 0; i < ITERS; ++i) {
        // streamed-once data: non-temporal 128-bit load (global_load_b128, TH=NT)
        v4f v = __builtin_nontemporal_load(&p[t + i * TPB]);
        s0 += v.x; s1 += v.y; s2 += v.z; s3 += v.w;
        q0 = fmaf(v.x, v.x, q0);
        q1 = fmaf(v.y, v.y, q1);
        q2 = fmaf(v.z, v.z, q2);
        q3 = fmaf(v.w, v.w, q3);
    }

    float s = (s0 + s1) + (s2 + s3);
    float q = (q0 + q1) + (q2 + q3);

    // wave32 butterfly reduction (gfx1250 is wave32-only: width 32, offsets 16..1)
#pragma unroll
    for (int off = 16; off > 0; off >>= 1) {
        s += __shfl_xor(s, off, 32);
        q += __shfl_xor(q, off, 32);
    }

    __shared__ float ls[TPB / 32];
    __shared__ float lq[TPB / 32];
    const int wave = t >> 5;
    if ((t & 31) == 0) { ls[wave] = s; lq[wave] = q; }
    __syncthreads();

    if (t == 0) {
        float S = ls[0], Q = lq[0];
#pragma unroll
        for (int w = 1; w < TPB / 32; ++w) { S += ls[w]; Q += lq[w]; }
        // final combine in f64: kills the S^2/N cancellation error; 1024 total
        const double N = (double)HW;
        double var = ((double)Q - (double)S * (double)S / N) / (N - 1.0);
        vars[slice] = (float)var;
    }
}

// Kernel 2: one wave (EXEC all-1s, as WMMA requires). Reduce the 1024 slice
// variances with V_WMMA_F32_16X16X4_F32: A = ones(16x4), B(4x16) carries the
// 64 per-lane fp32 partials (2 per lane), so D[0][n] = column-sums of B.
// Row 0 of D lives in d[0] across lanes 0..15 (C/D layout); a 16-lane
// butterfly gives the grand total. Fully fp32-exact use of the matrix unit.
__global__ __launch_bounds__(32)
void final_reduce_kernel(const float* __restrict__ vars, float* __restrict__ out)
{
    const int lane = threadIdx.x;

    v2f b = {0.f, 0.f};
#pragma unroll
    for (int i = 0; i < 16; ++i)  b.x += vars[lane + 32 * i];        // vars[0..511]
#pragma unroll
    for (int i = 16; i < 32; ++i) b.y += vars[lane + 32 * i];        // vars[512..1023]

    const v2f a = {1.0f, 1.0f};   // A = ones(16x4)
    v8f c = {};
    // 8 args: (neg_a, A, neg_b, B, c_mod, C, reuse_a, reuse_b)
    c = __builtin_amdgcn_wmma_f32_16x16x4_f32(false, a, false, b,
                                              (short)0, c, false, false);

    float x = c[0];               // row M=0 of D: lanes 0..15 hold the 16 column sums
    x += __shfl_xor(x, 8, 32);    // butterfly within the 16-lane group
    x += __shfl_xor(x, 4, 32);
    x += __shfl_xor(x, 2, 32);
    x += __shfl_xor(x, 1, 32);

    if (lane == 0) out[0] = sqrtf(x);
}

extern "C" void kernel_launch(void* const* d_in, const int* in_sizes, int n_in,
                              void* d_out, int out_size, void* d_ws, size_t ws_size,
                              hipStream_t stream)
{
    (void)in_sizes; (void)n_in; (void)out_size; (void)ws_size;
    const float* softmask = (const float*)d_in[0];   // [16,64,240,320] fp32
    float*       vars     = (float*)d_ws;            // 1024 per-slice variances (4 KB)
    float*       out      = (float*)d_out;           // scalar fp32

    slice_var_kernel<<<NSLICE, TPB, 0, stream>>>(softmask, vars);
    final_reduce_kernel<<<1, 32, 0, stream>>>(vars, out);
}